// Voxelization_85358180040888
// MI455X (gfx1250) — compile-verified
//
#include <hip/hip_runtime.h>

#define B_   8
#define C_   64
#define N_   100000
#define R_   32
#define R3_  32768
#define NB_  128     // partial-reduction blocks per batch
#define TPB  256

// workspace layout (float offsets)
#define OFF_PS    0                        // [B][NB][3] partial sums
#define OFF_MEAN  (OFF_PS + B_*NB_*3)      // [B][3]
#define OFF_PM    (OFF_MEAN + B_*3)        // [B][NB] partial max r^2
#define OFF_SCALE (OFF_PM + B_*NB_)        // [B]
#define OFF_CNT   (OFF_SCALE + B_)         // [B][R3] counts

typedef unsigned u32x4 __attribute__((ext_vector_type(4)));
typedef unsigned u32x8 __attribute__((ext_vector_type(8)));

__global__ void k_zero4(float4* __restrict__ p, int n4) {
  int i = blockIdx.x * blockDim.x + threadIdx.x;
  if (i < n4) p[i] = make_float4(0.f, 0.f, 0.f, 0.f);
}

__global__ void k_mean_partial(const float* __restrict__ coords, float* __restrict__ ws) {
  int b = blockIdx.x / NB_, blk = blockIdx.x % NB_, tid = threadIdx.x;
  const float* cb = coords + (size_t)b * 3 * N_;
  float sx = 0.f, sy = 0.f, sz = 0.f;
  for (int n = blk * TPB + tid; n < N_; n += NB_ * TPB) {
    sx += cb[n];
    sy += cb[N_ + n];
    sz += cb[2 * N_ + n];
  }
  __shared__ float sm[3][TPB];
  sm[0][tid] = sx; sm[1][tid] = sy; sm[2][tid] = sz;
  __syncthreads();
  for (int s = TPB / 2; s > 0; s >>= 1) {
    if (tid < s) {
      sm[0][tid] += sm[0][tid + s];
      sm[1][tid] += sm[1][tid + s];
      sm[2][tid] += sm[2][tid + s];
    }
    __syncthreads();
  }
  if (tid == 0) {
    float* ps = ws + OFF_PS + ((size_t)b * NB_ + blk) * 3;
    ps[0] = sm[0][0]; ps[1] = sm[1][0]; ps[2] = sm[2][0];
  }
}

__global__ void k_mean_final(float* __restrict__ ws) {
  int b = blockIdx.x, tid = threadIdx.x;
  const float* ps = ws + OFF_PS + (size_t)b * NB_ * 3;
  float x = 0.f, y = 0.f, z = 0.f;
  if (tid < NB_) { x = ps[tid * 3]; y = ps[tid * 3 + 1]; z = ps[tid * 3 + 2]; }
  __shared__ float sm[3][TPB];
  sm[0][tid] = x; sm[1][tid] = y; sm[2][tid] = z;
  __syncthreads();
  for (int s = TPB / 2; s > 0; s >>= 1) {
    if (tid < s) {
      sm[0][tid] += sm[0][tid + s];
      sm[1][tid] += sm[1][tid + s];
      sm[2][tid] += sm[2][tid + s];
    }
    __syncthreads();
  }
  if (tid == 0) {
    float invn = 1.0f / (float)N_;
    float* m = ws + OFF_MEAN + b * 3;
    m[0] = sm[0][0] * invn; m[1] = sm[1][0] * invn; m[2] = sm[2][0] * invn;
  }
}

__global__ void k_rad_partial(const float* __restrict__ coords, float* __restrict__ ws) {
  int b = blockIdx.x / NB_, blk = blockIdx.x % NB_, tid = threadIdx.x;
  const float* cb = coords + (size_t)b * 3 * N_;
  const float* m = ws + OFF_MEAN + b * 3;
  float mx = m[0], my = m[1], mz = m[2];
  float mr = 0.f;
  for (int n = blk * TPB + tid; n < N_; n += NB_ * TPB) {
    float dx = cb[n] - mx, dy = cb[N_ + n] - my, dz = cb[2 * N_ + n] - mz;
    mr = fmaxf(mr, dx * dx + dy * dy + dz * dz);
  }
  __shared__ float sm[TPB];
  sm[tid] = mr;
  __syncthreads();
  for (int s = TPB / 2; s > 0; s >>= 1) {
    if (tid < s) sm[tid] = fmaxf(sm[tid], sm[tid + s]);
    __syncthreads();
  }
  if (tid == 0) ws[OFF_PM + (size_t)b * NB_ + blk] = sm[0];
}

__global__ void k_rad_final(float* __restrict__ ws) {
  int b = blockIdx.x, tid = threadIdx.x;
  float v = (tid < NB_) ? ws[OFF_PM + (size_t)b * NB_ + tid] : 0.f;
  __shared__ float sm[TPB];
  sm[tid] = v;
  __syncthreads();
  for (int s = TPB / 2; s > 0; s >>= 1) {
    if (tid < s) sm[tid] = fmaxf(sm[tid], sm[tid + s]);
    __syncthreads();
  }
  if (tid == 0) ws[OFF_SCALE + b] = 2.0f * sqrtf(sm[0]);   // EPS = 0
}

__global__ void __launch_bounds__(TPB)
k_scatter(const float* __restrict__ features,
          const float* __restrict__ coords,
          const float* __restrict__ ws,
          float* __restrict__ out,
          float* __restrict__ counts) {
  const int nb = (N_ + TPB - 1) / TPB;
  int b = blockIdx.x / nb, blk = blockIdx.x % nb, tid = threadIdx.x;
  int n0 = blk * TPB;
  int n = n0 + tid;
  bool act = (n < N_);
  int ncl = act ? n : (N_ - 1);       // clamp so every lane issues a valid address
  const int tw = (N_ - n0 < TPB) ? (N_ - n0) : TPB;  // tile width (tail-safe)

  __shared__ float sf[C_ * TPB];      // 64KB: features tile [C][tw] row-packed
  __shared__ float sc[3][TPB];        // coord staging

  // ---- TDM: one DMA per block pulls features[b, 0:64, n0:n0+tw] into LDS ----
  if (tid < 32) {
    unsigned long long gaddr =
        (unsigned long long)(features + (size_t)b * C_ * N_ + (size_t)n0);
    unsigned lds_addr = (unsigned)(size_t)(&sf[0]);
    unsigned g0w0 = 1u;                                   // count=1
    unsigned g0w1 = lds_addr;
    unsigned g0w2 = (unsigned)gaddr;
    unsigned g0w3 = ((unsigned)(gaddr >> 32) & 0x01FFFFFFu) | (2u << 30); // type=2
    unsigned td0 = (unsigned)tw, td1 = (unsigned)C_;
    unsigned g1w0 = 2u << 16;                             // data_size=4B
    unsigned g1w1 = (td0 & 0xFFFFu) << 16;                // tensor_dim0[15:0]
    unsigned g1w2 = (td0 >> 16) | ((td1 & 0xFFFFu) << 16);// tensor_dim0[31:16] | tensor_dim1[15:0]
    unsigned g1w3 = (td1 >> 16) | ((unsigned)tw << 16);   // tensor_dim1[31:16] | tile_dim0
    unsigned g1w4 = (unsigned)C_;                         // tile_dim1=64, tile_dim2=0
    unsigned g1w5 = (unsigned)N_;                         // tensor_dim0_stride[31:0]
    unsigned g1w6 = 0u;                                   // stride0[47:32] | stride1[15:0]
    unsigned g1w7 = 0u;                                   // stride1[47:16]
    u32x4 g0; u32x8 g1;
    g0.x = __builtin_amdgcn_readfirstlane(g0w0);
    g0.y = __builtin_amdgcn_readfirstlane(g0w1);
    g0.z = __builtin_amdgcn_readfirstlane(g0w2);
    g0.w = __builtin_amdgcn_readfirstlane(g0w3);
    g1.s0 = __builtin_amdgcn_readfirstlane(g1w0);
    g1.s1 = __builtin_amdgcn_readfirstlane(g1w1);
    g1.s2 = __builtin_amdgcn_readfirstlane(g1w2);
    g1.s3 = __builtin_amdgcn_readfirstlane(g1w3);
    g1.s4 = __builtin_amdgcn_readfirstlane(g1w4);
    g1.s5 = __builtin_amdgcn_readfirstlane(g1w5);
    g1.s6 = __builtin_amdgcn_readfirstlane(g1w6);
    g1.s7 = __builtin_amdgcn_readfirstlane(g1w7);
    asm volatile("tensor_load_to_lds %0, %1" :: "s"(g0), "s"(g1) : "memory");
  }

  // ---- async global->LDS stage of the 3 strided coord streams (all waves) ----
  unsigned l0 = (unsigned)(size_t)(&sc[0][tid]);
  unsigned l1 = (unsigned)(size_t)(&sc[1][tid]);
  unsigned l2 = (unsigned)(size_t)(&sc[2][tid]);
  unsigned cg0 = (unsigned)(((size_t)b * 3 * N_ + (size_t)ncl) * 4u);
  unsigned cg1 = cg0 + (unsigned)(N_ * 4u);
  unsigned cg2 = cg1 + (unsigned)(N_ * 4u);
  asm volatile("global_load_async_to_lds_b32 %0, %1, %2"
               :: "v"(l0), "v"(cg0), "s"(coords) : "memory");
  asm volatile("global_load_async_to_lds_b32 %0, %1, %2"
               :: "v"(l1), "v"(cg1), "s"(coords) : "memory");
  asm volatile("global_load_async_to_lds_b32 %0, %1, %2"
               :: "v"(l2), "v"(cg2), "s"(coords) : "memory");

  // batch-uniform params (scalar loads) overlap with the DMA
  const float* m = ws + OFF_MEAN + b * 3;
  float mx = m[0], my = m[1], mz = m[2];
  float inv = 1.0f / ws[OFF_SCALE + b];

  asm volatile("s_wait_asynccnt 0x0" ::: "memory");
  float x = sc[0][tid], y = sc[1][tid], z = sc[2][tid];

  float fx = fminf(fmaxf(((x - mx) * inv + 0.5f) * (float)R_, 0.f), (float)(R_ - 1));
  float fy = fminf(fmaxf(((y - my) * inv + 0.5f) * (float)R_, 0.f), (float)(R_ - 1));
  float fz = fminf(fmaxf(((z - mz) * inv + 0.5f) * (float)R_, 0.f), (float)(R_ - 1));

  // wave 0 waits for the TDM tile, then the whole block syncs on it
  if (tid < 32) __builtin_amdgcn_s_wait_tensorcnt(0);
  __syncthreads();

  if (act) {
    // norm_coords output (second tuple element), laid out after vox_feat
    float* nco = out + (size_t)B_ * C_ * R3_ + (size_t)b * 3 * N_;
    nco[n] = fx; nco[N_ + n] = fy; nco[2 * N_ + n] = fz;

    int vx = (int)rintf(fx), vy = (int)rintf(fy), vz = (int)rintf(fz);  // round half-even == jnp.round
    int idx = (vx * R_ + vy) * R_ + vz;

    unsafeAtomicAdd(&counts[(size_t)b * R3_ + idx], 1.0f);

    float* dst = out + (size_t)b * C_ * R3_ + idx;   // stride R3_ over channels
    #pragma unroll 8
    for (int c = 0; c < C_; ++c)
      unsafeAtomicAdd(dst + (size_t)c * R3_, sf[c * tw + tid]);  // LDS-fed, conflict-free
  }
}

__global__ void k_finalize(float* __restrict__ out, const float* __restrict__ counts) {
  size_t i = (size_t)blockIdx.x * TPB + threadIdx.x;   // over B*C*R3
  int v = (int)(i & (R3_ - 1));
  int b = (int)(i >> 15) >> 6;
  float d = fmaxf(counts[(size_t)b * R3_ + v], 1.0f);
  out[i] = out[i] / d;
}

extern "C" void kernel_launch(void* const* d_in, const int* in_sizes, int n_in,
                              void* d_out, int out_size, void* d_ws, size_t ws_size,
                              hipStream_t stream) {
  const float* features = (const float*)d_in[0];
  const float* coords   = (const float*)d_in[1];
  float* out = (float*)d_out;
  float* ws  = (float*)d_ws;
  float* counts = ws + OFF_CNT;
  (void)in_sizes; (void)n_in; (void)out_size; (void)ws_size;

  // zero voxel accumulator region of d_out and the counts
  {
    int n4 = (B_ * C_ * R3_) / 4;
    k_zero4<<<n4 / TPB, TPB, 0, stream>>>((float4*)out, n4);
    int c4 = (B_ * R3_) / 4;
    k_zero4<<<(c4 + TPB - 1) / TPB, TPB, 0, stream>>>((float4*)counts, c4);
  }

  k_mean_partial<<<B_ * NB_, TPB, 0, stream>>>(coords, ws);
  k_mean_final  <<<B_, TPB, 0, stream>>>(ws);
  k_rad_partial <<<B_ * NB_, TPB, 0, stream>>>(coords, ws);
  k_rad_final   <<<B_, TPB, 0, stream>>>(ws);

  int nb = (N_ + TPB - 1) / TPB;
  k_scatter<<<B_ * nb, TPB, 0, stream>>>(features, coords, ws, out, counts);

  k_finalize<<<(B_ * C_ * R3_) / TPB, TPB, 0, stream>>>(out, counts);
}